// GPSLayer_29085518528618
// MI455X (gfx1250) — compile-verified
//
#include <hip/hip_runtime.h>
#include <math.h>

// ---------------------------------------------------------------------------
// GPS graph-attention layer for MI455X (gfx1250, wave32, WMMA).
//   N=50000 nodes, E=800000 edges, IN_DIM=128, H=8 heads, D=16 (d=128).
// ---------------------------------------------------------------------------

typedef __attribute__((ext_vector_type(2))) float v2f;
typedef __attribute__((ext_vector_type(8))) float v8f;

#define IN_DIM 128
#define DMODEL 128
#define NHEAD  8
#define HDIM   16

// ---------------------------------------------------------------------------
// Kernel 1: P = softmax(S, axis=1).  S is tiny (4 x 64) -> one wave.
// ---------------------------------------------------------------------------
__global__ void softmaxS_kernel(const float* __restrict__ S, float* __restrict__ P) {
    int r = threadIdx.x;
    if (r < 4) {
        float m = -1e30f;
        for (int j = 0; j < 64; ++j) m = fmaxf(m, S[r * 64 + j]);
        float sum = 0.f;
        for (int j = 0; j < 64; ++j) {
            float e = expf(S[r * 64 + j] - m);
            P[r * 64 + j] = e;
            sum += e;
        }
        float inv = 1.f / sum;
        for (int j = 0; j < 64; ++j) P[r * 64 + j] *= inv;
    }
}

// ---------------------------------------------------------------------------
// Kernel 2: fused QKV GEMM (V_WMMA_F32_16X16X4_F32) + bias + RoPE rotation.
//   Block = 256 threads = 8 waves; block owns a 16-node row strip.
//   Wave w owns head w (output cols [16w,16w+16)) for Q, K, V.
//   K-loop: 32 steps of K=4, 3 WMMAs per step (Q,K,V) -> 96 wmma per wave.
//
//   fp32 A-frag layout (16x4, ISA 7.12.2): lanes 0-15 & 16-31 both hold rows
//   M=lane&15; VGPR v holds K = kk + 2*(lane>>4) + v.  B-frag (4x16) mirrors
//   this: lane holds col N=lane&15, VGPR v holds row K = kk + 2*(lane>>4) + v.
//   C/D (16x16): VGPR r = row r (lanes 0-15) / row r+8 (lanes 16-31), N=lane&15.
//
//   RoPE: head h occupies exactly one 16-col tile, so the rot2 partner column
//   is another lane of the SAME wave -> __shfl (wave32).  theta index per lane
//   is fixed (h*8 + col/2); theta per row = dot(angles[row], P[:,idx]).
// ---------------------------------------------------------------------------
__global__ __launch_bounds__(256)
void qkv_wmma_rope_kernel(const float* __restrict__ x,
                          const float* __restrict__ angles,
                          const float* __restrict__ Wq, const float* __restrict__ bq,
                          const float* __restrict__ Wk, const float* __restrict__ bk,
                          const float* __restrict__ Wv, const float* __restrict__ bv,
                          const float* __restrict__ P,
                          float* __restrict__ Qo, float* __restrict__ Ko,
                          float* __restrict__ Vo, int nNodes)
{
    const int lane  = threadIdx.x & 31;
    const int head  = threadIdx.x >> 5;      // 0..7
    const int node0 = blockIdx.x * 16;
    const int half  = lane >> 4;             // 0 or 1
    const int ln    = lane & 15;
    const int colBase = head * 16;
    const int col   = colBase + ln;

    v8f cq = {}; v8f ck = {}; v8f cv = {};

    int arow = node0 + ln;
    if (arow >= nNodes) arow = nNodes - 1;   // clamp (no divergence: WMMA needs EXEC all-1)
    const float* xrow = x + (size_t)arow * IN_DIM;

    for (int kk = 0; kk < IN_DIM; kk += 4) {
        const int k0 = kk + half * 2;
        v2f a; a.x = xrow[k0]; a.y = xrow[k0 + 1];

        v2f b;
        b.x = Wq[(size_t)k0 * DMODEL + col];
        b.y = Wq[(size_t)(k0 + 1) * DMODEL + col];
        cq = __builtin_amdgcn_wmma_f32_16x16x4_f32(false, a, false, b, (short)0, cq, false, false);

        b.x = Wk[(size_t)k0 * DMODEL + col];
        b.y = Wk[(size_t)(k0 + 1) * DMODEL + col];
        ck = __builtin_amdgcn_wmma_f32_16x16x4_f32(false, a, false, b, (short)0, ck, false, false);

        b.x = Wv[(size_t)k0 * DMODEL + col];
        b.y = Wv[(size_t)(k0 + 1) * DMODEL + col];
        cv = __builtin_amdgcn_wmma_f32_16x16x4_f32(false, a, false, b, (short)0, cv, false, false);
    }

    const float bqv = bq[col], bkv = bk[col], bvv = bv[col];

    // RoPE parameters fixed per lane (column fixed): theta idx = h*8 + j/2
    const int j = ln;
    const int thetaIdx = colBase / 2 + (j >> 1);
    const float p0 = P[0 * 64 + thetaIdx];
    const float p1 = P[1 * 64 + thetaIdx];
    const float p2 = P[2 * 64 + thetaIdx];
    const float p3 = P[3 * 64 + thetaIdx];
    // rot2 within a head: out[j] = (j<8) ? -q[2j+1] : q[2(j-8)]
    const int   pj  = (j < 8) ? (2 * j + 1) : (2 * (j - 8));
    const float sgn = (j < 8) ? -1.f : 1.f;
    const int   partnerLane = half * 16 + pj;  // same row-half, partner column

#pragma unroll
    for (int r = 0; r < 8; ++r) {
        float qv = cq[r] + bqv;
        float kv = ck[r] + bkv;
        float vv = cv[r] + bvv;
        // shuffles are executed by all lanes (uniform control flow)
        float qpart = __shfl(qv, partnerLane, 32);
        float kpart = __shfl(kv, partnerLane, 32);
        int row = node0 + r + half * 8;
        if (row < nNodes) {
            const float4 ang = *(const float4*)(angles + (size_t)row * 4);
            float th = ang.x * p0 + ang.y * p1 + ang.z * p2 + ang.w * p3;
            float c = cosf(th), s = sinf(th);
            size_t off = (size_t)row * DMODEL + col;
            Qo[off] = qv * c + sgn * qpart * s;
            Ko[off] = kv * c + sgn * kpart * s;
            Vo[off] = vv;
        }
    }
}

// ---------------------------------------------------------------------------
// Kernel 3: per-(edge,head) score = clamp(dot(K[src],Q[dst])/4, +-5);
// exp-score stored, atomic per-(dst,head) sum.  Scores clamped to +-5 =>
// exp() cannot overflow: the reference's segment-max subtraction cancels
// mathematically, so we skip that entire extra pass.
// ---------------------------------------------------------------------------
__global__ __launch_bounds__(256)
void edge_score_kernel(const int* __restrict__ ei,
                       const float* __restrict__ Q, const float* __restrict__ K,
                       float* __restrict__ expsc, float* __restrict__ ssum, int E)
{
    int t = blockIdx.x * blockDim.x + threadIdx.x;
    if (t >= E * NHEAD) return;
    int e = t >> 3, h = t & 7;
    int src = ei[e];
    int dst = ei[E + e];
    const float4* kp = (const float4*)(K + (size_t)src * DMODEL + h * HDIM);
    const float4* qp = (const float4*)(Q + (size_t)dst * DMODEL + h * HDIM);
    float acc = 0.f;
#pragma unroll
    for (int i = 0; i < 4; ++i) {
        float4 kv = kp[i], qv = qp[i];
        acc += kv.x * qv.x + kv.y * qv.y + kv.z * qv.z + kv.w * qv.w;
    }
    acc *= 0.25f;                               // 1/sqrt(16)
    acc = fminf(5.f, fmaxf(-5.f, acc));         // CLAMP
    float ex = expf(acc);
    expsc[t] = ex;
    atomicAdd(&ssum[(size_t)dst * NHEAD + h], ex);
}

// ---------------------------------------------------------------------------
// Kernel 4: normalize + scatter-add messages:  out[dst,h,:] += V[src,h,:]*a
// ---------------------------------------------------------------------------
__global__ __launch_bounds__(256)
void edge_aggregate_kernel(const int* __restrict__ ei,
                           const float* __restrict__ V,
                           const float* __restrict__ expsc,
                           const float* __restrict__ ssum,
                           float* __restrict__ out, int E)
{
    int t = blockIdx.x * blockDim.x + threadIdx.x;
    if (t >= E * NHEAD) return;
    int e = t >> 3, h = t & 7;
    int src = ei[e];
    int dst = ei[E + e];
    float coeff = expsc[t] / (ssum[(size_t)dst * NHEAD + h] + 1e-16f);
    const float4* vp = (const float4*)(V + (size_t)src * DMODEL + h * HDIM);
    float* op = out + (size_t)dst * DMODEL + h * HDIM;
#pragma unroll
    for (int i = 0; i < 4; ++i) {
        float4 v = vp[i];
        atomicAdd(op + 4 * i + 0, v.x * coeff);
        atomicAdd(op + 4 * i + 1, v.y * coeff);
        atomicAdd(op + 4 * i + 2, v.z * coeff);
        atomicAdd(op + 4 * i + 3, v.w * coeff);
    }
}

// ---------------------------------------------------------------------------
// Launch
// ---------------------------------------------------------------------------
extern "C" void kernel_launch(void* const* d_in, const int* in_sizes, int n_in,
                              void* d_out, int out_size, void* d_ws, size_t ws_size,
                              hipStream_t stream) {
    const float* x      = (const float*)d_in[0];
    const float* angles = (const float*)d_in[1];
    const int*   ei     = (const int*)  d_in[2];
    const float* Wq     = (const float*)d_in[3];
    const float* bq     = (const float*)d_in[4];
    const float* Wk     = (const float*)d_in[5];
    const float* bk     = (const float*)d_in[6];
    const float* Wv     = (const float*)d_in[7];
    const float* bv     = (const float*)d_in[8];
    const float* S      = (const float*)d_in[9];

    const int N = in_sizes[0] / IN_DIM;
    const int E = in_sizes[2] / 2;

    // workspace layout (floats):  P | Q | K | V | expsc | ssum   (~104 MB)
    float* ws    = (float*)d_ws;
    float* P     = ws;                         // 256
    float* Q     = ws + 256;                   // N*128
    float* K     = Q + (size_t)N * DMODEL;     // N*128
    float* V     = K + (size_t)N * DMODEL;     // N*128
    float* expsc = V + (size_t)N * DMODEL;     // E*8
    float* ssum  = expsc + (size_t)E * NHEAD;  // N*8

    hipMemsetAsync(d_out, 0, (size_t)out_size * sizeof(float), stream);
    hipMemsetAsync(ssum, 0, (size_t)N * NHEAD * sizeof(float), stream);

    softmaxS_kernel<<<1, 32, 0, stream>>>(S, P);

    qkv_wmma_rope_kernel<<<(N + 15) / 16, 256, 0, stream>>>(
        x, angles, Wq, bq, Wk, bk, Wv, bv, P, Q, K, V, N);

    int tot = E * NHEAD;
    edge_score_kernel<<<(tot + 255) / 256, 256, 0, stream>>>(ei, Q, K, expsc, ssum, E);
    edge_aggregate_kernel<<<(tot + 255) / 256, 256, 0, stream>>>(ei, V, expsc, ssum,
                                                                 (float*)d_out, E);
}